// RouterSinkhorn_22995254902962
// MI455X (gfx1250) — compile-verified
//
#include <hip/hip_runtime.h>
#include <hip/hip_bf16.h>

typedef __attribute__((ext_vector_type(2))) float v2f;
typedef __attribute__((ext_vector_type(8))) float v8f;

// Check only in the device pass (__HIP_DEVICE_COMPILE__ is set only there).
#if defined(__HIP_DEVICE_COMPILE__) && !__has_builtin(__builtin_amdgcn_wmma_f32_16x16x4_f32)
#error "missing __builtin_amdgcn_wmma_f32_16x16x4_f32 (device pass)"
#endif

#define T_TOK 8192
#define HID   2048
#define NEXP  64
#define EPS_F 1e-8f

#define BM  128
#define BK  32
#define LDA 36   // padded LDS row stride (floats): conflict-free & 16B-store aligned

// ---------------------------------------------------------------------------
// GEMM: logits = hidden(8192x2048) @ W^T(2048x64), fp32 WMMA 16x16x4.
// Also emits sigmoid(logits) and exp(logits) (Sinkhorn cost) in the epilogue.
// ---------------------------------------------------------------------------
__global__ __launch_bounds__(256) void router_gemm_wmma(
    const float* __restrict__ hidden, const float* __restrict__ W,
    float* __restrict__ logits, float* __restrict__ affin,
    float* __restrict__ cost)
{
  __shared__ float As[BM * LDA];    // 18432 B
  __shared__ float Bs[NEXP * LDA];  //  9216 B

  const int tid   = threadIdx.x;
  const int wave  = tid >> 5;       // 0..7
  const int lane  = tid & 31;
  const int l16   = lane & 15;
  const int half  = lane >> 4;      // 0/1 selects K pair within fragment
  const int tbase = blockIdx.x * BM;

  const int ldRow = tid >> 3;        // 0..31
  const int ldCol = (tid & 7) * 4;   // 0..28 step 4

  v8f c0 = {}, c1 = {}, c2 = {}, c3 = {};

  for (int k0 = 0; k0 < HID; k0 += BK) {
    // ---- stage A tile: 128 x 32 (coalesced float4) ----
#pragma unroll
    for (int r = 0; r < 4; ++r) {
      const int row = r * 32 + ldRow;
      const float4 v = *(const float4*)(hidden + (size_t)(tbase + row) * HID + k0 + ldCol);
      *(float4*)&As[row * LDA + ldCol] = v;
    }
    // ---- stage B tile: 64 x 32 ----
#pragma unroll
    for (int r = 0; r < 2; ++r) {
      const int row = r * 32 + ldRow;
      const float4 v = *(const float4*)(W + (size_t)row * HID + k0 + ldCol);
      *(float4*)&Bs[row * LDA + ldCol] = v;
    }
    // prefetch next A chunk into cache hierarchy (gfx1250 global_prefetch)
    if (k0 + BK < HID)
      __builtin_prefetch(hidden + (size_t)(tbase + ldRow) * HID + (k0 + BK) + ldCol, 0, 0);
    __syncthreads();

#if defined(__HIP_DEVICE_COMPILE__)
#pragma unroll
    for (int kk = 0; kk < BK; kk += 4) {
      const v2f a  = *(const v2f*)&As[(wave * 16 + l16) * LDA + kk + 2 * half];
      const v2f b0 = *(const v2f*)&Bs[( 0 + l16) * LDA + kk + 2 * half];
      const v2f b1 = *(const v2f*)&Bs[(16 + l16) * LDA + kk + 2 * half];
      const v2f b2 = *(const v2f*)&Bs[(32 + l16) * LDA + kk + 2 * half];
      const v2f b3 = *(const v2f*)&Bs[(48 + l16) * LDA + kk + 2 * half];
      c0 = __builtin_amdgcn_wmma_f32_16x16x4_f32(false, a, false, b0, (short)0, c0, false, false);
      c1 = __builtin_amdgcn_wmma_f32_16x16x4_f32(false, a, false, b1, (short)0, c1, false, false);
      c2 = __builtin_amdgcn_wmma_f32_16x16x4_f32(false, a, false, b2, (short)0, c2, false, false);
      c3 = __builtin_amdgcn_wmma_f32_16x16x4_f32(false, a, false, b3, (short)0, c3, false, false);
    }
#endif
    __syncthreads();
  }

  // ---- epilogue: C layout (lane = N, VGPR r -> M = r + 8*half) ----
#pragma unroll
  for (int r = 0; r < 8; ++r) {
    const int t = tbase + wave * 16 + half * 8 + r;
    const size_t base = (size_t)t * NEXP + l16;
    const float v[4] = { c0[r], c1[r], c2[r], c3[r] };
#pragma unroll
    for (int j = 0; j < 4; ++j) {
      const float x = v[j];
      const size_t o = base + j * 16;
      logits[o] = x;
      affin[o]  = 1.0f / (1.0f + __expf(-x));
      cost[o]   = __expf(x);
    }
  }
}

// ---------------------------------------------------------------------------
// Sinkhorn: state is d1[64] only. Each pass fuses d0 computation with the
// per-expert column-sum accumulation (one sweep over the L2-resident cost).
// ---------------------------------------------------------------------------
__global__ void sinkhorn_init(float* __restrict__ d1, float* __restrict__ esum)
{
  const int e = threadIdx.x;
  d1[e] = 1.0f;
  esum[e] = 0.0f;
}

__global__ __launch_bounds__(256) void sinkhorn_pass(
    const float* __restrict__ cost, const float* __restrict__ d1,
    float* __restrict__ esum)
{
  __shared__ float sd1[NEXP];
  __shared__ float part[NEXP];
  const int tid = threadIdx.x;
  if (tid < NEXP) { sd1[tid] = d1[tid]; part[tid] = 0.0f; }
  __syncthreads();

  const int t = blockIdx.x * 256 + tid;
  const float4* row4 = (const float4*)(cost + (size_t)t * NEXP);

  float4 rv[16];
  float s = 0.0f;
#pragma unroll
  for (int i = 0; i < 16; ++i) {
    rv[i] = row4[i];
    s += rv[i].x * sd1[4 * i + 0] + rv[i].y * sd1[4 * i + 1] +
         rv[i].z * sd1[4 * i + 2] + rv[i].w * sd1[4 * i + 3];
  }
  const float d0 = (1.0f / (float)T_TOK) / (s + EPS_F);
#pragma unroll
  for (int i = 0; i < 16; ++i) {
    atomicAdd(&part[4 * i + 0], d0 * rv[i].x);
    atomicAdd(&part[4 * i + 1], d0 * rv[i].y);
    atomicAdd(&part[4 * i + 2], d0 * rv[i].z);
    atomicAdd(&part[4 * i + 3], d0 * rv[i].w);
  }
  __syncthreads();
  if (tid < NEXP) atomicAdd(&esum[tid], part[tid]);
}

__global__ void sinkhorn_update(float* __restrict__ d1, float* __restrict__ esum)
{
  const int e = threadIdx.x;
  d1[e] = (1.0f / (float)NEXP) / (esum[e] + EPS_F);
  esum[e] = 0.0f;   // ready for next pass
}

// argmax_e( d1[e] * cost[t,e] * d0[t] ) == argmax_e( d1[e] * cost[t,e] )
__global__ __launch_bounds__(256) void sinkhorn_argmax(
    const float* __restrict__ cost, const float* __restrict__ d1,
    float* __restrict__ idx_out)
{
  __shared__ float sd1[NEXP];
  const int tid = threadIdx.x;
  if (tid < NEXP) sd1[tid] = d1[tid];
  __syncthreads();

  const int t = blockIdx.x * 256 + tid;
  const float* row = cost + (size_t)t * NEXP;
  float best = -1.0f;
  int bi = 0;
#pragma unroll
  for (int e = 0; e < NEXP; ++e) {
    const float v = sd1[e] * row[e];
    if (v > best) { best = v; bi = e; }   // first occurrence wins (jnp.argmax)
  }
  idx_out[t] = (float)bi;
}

// ---------------------------------------------------------------------------
extern "C" void kernel_launch(void* const* d_in, const int* in_sizes, int n_in,
                              void* d_out, int out_size, void* d_ws, size_t ws_size,
                              hipStream_t stream) {
  const float* hidden = (const float*)d_in[0];   // 4*2048*2048 fp32
  const float* W      = (const float*)d_in[1];   // 64*2048 fp32

  float* out    = (float*)d_out;
  float* logits = out;                           // 8192*64
  float* affin  = out + (size_t)T_TOK * NEXP;    // 8192*64
  float* idxout = out + 2 * (size_t)T_TOK * NEXP;// 8192

  float* cost = (float*)d_ws;                    // 8192*64 fp32 (2 MiB)
  float* d1   = cost + (size_t)T_TOK * NEXP;     // 64
  float* esum = d1 + NEXP;                       // 64

  router_gemm_wmma<<<T_TOK / BM, 256, 0, stream>>>(hidden, W, logits, affin, cost);
  sinkhorn_init<<<1, NEXP, 0, stream>>>(d1, esum);
  for (int it = 0; it < 30; ++it) {
    sinkhorn_pass<<<T_TOK / 256, 256, 0, stream>>>(cost, d1, esum);
    sinkhorn_update<<<1, NEXP, 0, stream>>>(d1, esum);
  }
  sinkhorn_argmax<<<T_TOK / 256, 256, 0, stream>>>(cost, d1, idxout);
}